// SplitNN_4879082848949
// MI455X (gfx1250) — compile-verified
//
#include <hip/hip_runtime.h>

typedef __attribute__((ext_vector_type(2))) float v2f;
typedef __attribute__((ext_vector_type(8))) float v8f;

#define EPS_CONST (4.0f / 30.0f)

// One fused workgroup: 16 waves (wave32) = 512 threads.
// waves 0..15 -> (mtile = w>>3 in 0..1, ntile = w&7 in 0..7), each owns a
// 16x16 fp32 WMMA tile of both o1 = x1@W1+b1 and o2 = x2@W2+b2.
__global__ __launch_bounds__(512) void splitnn_fused(
    const float* __restrict__ x1,     // [32,160]
    const float* __restrict__ x2,     // [32,160]
    const float* __restrict__ impor,  // [1,128]
    const float* __restrict__ W1,     // [160,128]
    const float* __restrict__ b1,     // [128]
    const float* __restrict__ W2,     // [160,128]
    const float* __restrict__ b2,     // [128]
    const float* __restrict__ Ws,     // [128,2]
    const float* __restrict__ bs,     // [2]
    const float* __restrict__ lap,    // [32,128]
    const float* __restrict__ rr,     // [128]
    float* __restrict__ out)          // [32,2]
{
    __shared__ float o1s[32 * 128];
    __shared__ float o2s[32 * 128];
    __shared__ float nms[128];        // masked noise multiplier per column

    const int tid   = threadIdx.x;
    const int lane  = tid & 31;
    const int wave  = tid >> 5;       // 0..15
    const int mtile = wave >> 3;      // 0..1
    const int ntile = wave & 7;       // 0..7
    const int half  = lane >> 4;      // 0: K=k0,k0+1 ; 1: K=k0+2,k0+3
    const int l     = lane & 15;
    const int m     = mtile * 16 + l; // A-matrix row for this lane
    const int n     = ntile * 16 + l; // B-matrix column for this lane

    const float* x1r = x1 + m * 160;
    const float* x2r = x2 + m * 160;

    v8f acc1 = {};
    v8f acc2 = {};

    // K = 160 -> 40 steps of V_WMMA_F32_16X16X4_F32 per GEMM.
    // Two independent accumulator chains (o1/o2) hide WMMA RAW latency.
    #pragma unroll 4
    for (int k0 = 0; k0 < 160; k0 += 4) {
        const int ka = k0 + half * 2;
        v2f a, b;

        // A fragment (16x4 f32, ISA layout): lane half selects K pair
        a.x = x1r[ka];
        a.y = x1r[ka + 1];
        // B fragment (4x16 f32): lane = column, VGPR selects K within half
        b.x = W1[ka * 128 + n];
        b.y = W1[(ka + 1) * 128 + n];
        acc1 = __builtin_amdgcn_wmma_f32_16x16x4_f32(
            false, a, false, b, (short)0, acc1, false, false);

        a.x = x2r[ka];
        a.y = x2r[ka + 1];
        b.x = W2[ka * 128 + n];
        b.y = W2[(ka + 1) * 128 + n];
        acc2 = __builtin_amdgcn_wmma_f32_16x16x4_f32(
            false, a, false, b, (short)0, acc2, false, false);
    }

    // Scatter C tiles to LDS with bias add.
    // C/D layout: VGPR r -> (M = r + 8*half_of_lane, N = lane&15).
    const int col = ntile * 16 + l;
    #pragma unroll
    for (int r = 0; r < 8; ++r) {
        const int row = mtile * 16 + half * 8 + r;
        o1s[row * 128 + col] = acc1[r] + b1[col];
        o2s[row * 128 + col] = acc2[r] + b2[col];
    }
    __syncthreads();

    // Per-column sensitivity deltaF and masked noise multiplier (128 cols).
    if (tid < 128) {
        float mx = o1s[tid];
        float mn = mx;
        #pragma unroll
        for (int mm = 1; mm < 32; ++mm) {
            const float v = o1s[mm * 128 + tid];
            mx = fmaxf(mx, v);
            mn = fminf(mn, v);
        }
        const float dF = mx - mn;
        float es = impor[tid] * EPS_CONST;
        if (es == 0.0f) es = 1e-4f;
        float mul = dF / es;
        if (mul == 0.0f) mul = 1e-4f;
        nms[tid] = (dF == 0.0f) ? 0.0f : mul;   // noise zeroed where dF==0
    }
    __syncthreads();

    // DP noise + random-response mask on o1, then elementwise min fusion.
    for (int idx = tid; idx < 32 * 128; idx += 512) {
        const int nn = idx & 127;
        const float v1 = (o1s[idx] + lap[idx] * nms[nn]) * rr[nn];
        o1s[idx] = fminf(v1, o2s[idx]);         // server_in, in place
    }
    __syncthreads();

    // Final tiny GEMM: [32,128] @ [128,2] + bs -> 64 outputs, 1 thread each.
    if (tid < 64) {
        const int mm = tid >> 1;
        const int j  = tid & 1;
        float sum = bs[j];
        #pragma unroll 8
        for (int k = 0; k < 128; ++k)
            sum = fmaf(o1s[mm * 128 + k], Ws[k * 2 + j], sum);
        out[mm * 2 + j] = sum;
    }
}

extern "C" void kernel_launch(void* const* d_in, const int* in_sizes, int n_in,
                              void* d_out, int out_size, void* d_ws, size_t ws_size,
                              hipStream_t stream) {
    (void)in_sizes; (void)n_in; (void)d_ws; (void)ws_size; (void)out_size;
    const float* x1    = (const float*)d_in[0];
    const float* x2    = (const float*)d_in[1];
    const float* impor = (const float*)d_in[2];
    const float* W1    = (const float*)d_in[3];
    const float* b1    = (const float*)d_in[4];
    const float* W2    = (const float*)d_in[5];
    const float* b2    = (const float*)d_in[6];
    const float* Ws    = (const float*)d_in[7];
    const float* bs    = (const float*)d_in[8];
    const float* lap   = (const float*)d_in[9];
    const float* rr    = (const float*)d_in[10];
    float* out = (float*)d_out;

    splitnn_fused<<<1, 512, 0, stream>>>(x1, x2, impor, W1, b1, W2, b2,
                                         Ws, bs, lap, rr, out);
}